// MetaGNN_60765197304208
// MI455X (gfx1250) — compile-verified
//
#include <hip/hip_runtime.h>
#include <hip/hip_bf16.h>
#include <math.h>

#define N_NODES 8192
#define E_RAW   65536
#define D_EMB   256
#define H_HEADS 8
#define HD      32
#define E2      (2 * E_RAW + N_NODES)   // 139264, divisible by 16
#define KQV_LD  768
#define KSCALE  0.17677669529663687f    // 1/sqrt(32)

typedef float v2f __attribute__((ext_vector_type(2)));
typedef float v8f __attribute__((ext_vector_type(8)));

// D = A(16x4) * B(4x16) + C(16x16), fp32, wave32
#define WMMA_F32(a, b, c) \
    __builtin_amdgcn_wmma_f32_16x16x4_f32(false, (a), false, (b), (short)0, (c), false, false)

// ---------------------------------------------------------------- utilities
__device__ __forceinline__ void atomicMaxF(float* addr, float val) {
    unsigned int* u = (unsigned int*)addr;
    unsigned int old = __float_as_uint(*addr);
    while (__uint_as_float(old) < val) {
        unsigned int assumed = old;
        old = atomicCAS(u, assumed, __float_as_uint(val));
        if (old == assumed) break;
    }
}

// ---------------------------------------------------------------- 1) edges
__global__ void build_edges(const int* __restrict__ ei, const unsigned char* __restrict__ qm,
                            int* __restrict__ src, int* __restrict__ dst, int* __restrict__ valid) {
    int i = blockIdx.x * blockDim.x + threadIdx.x;
    if (i >= E2) return;
    if (i < E_RAW) {                           // support slots (valid iff !query_mask)
        src[i] = ei[i]; dst[i] = ei[E_RAW + i]; valid[i] = qm[i] ? 0 : 1;
    } else if (i < 2 * E_RAW) {                // flipped edges (all)
        int j = i - E_RAW;
        src[i] = ei[E_RAW + j]; dst[i] = ei[j]; valid[i] = 1;
    } else {                                   // self loops
        int n = i - 2 * E_RAW;
        src[i] = n; dst[i] = n; valid[i] = 1;
    }
}

// ---------------------------------------------------------------- 2) residual init
__global__ void init_out(const float4* __restrict__ x4, float4* __restrict__ out4) {
    int i = blockIdx.x * blockDim.x + threadIdx.x;
    if (i < N_NODES * D_EMB / 4) out4[i] = x4[i];
}

// ---------------------------------------------------------------- 3) kqv GEMM (WMMA)
// kqv[8192][768] = x[8192][256] @ W[256][768] + b. Wave computes 16 rows x 64 cols.
__global__ __launch_bounds__(256) void kqv_gemm(const float* __restrict__ x,
                                                const float* __restrict__ W,
                                                const float* __restrict__ b,
                                                float* __restrict__ kqv) {
    int lane = threadIdx.x & 31, wave = threadIdx.x >> 5;
    int idx = blockIdx.x * 8 + wave;           // 512 mtiles * 12 chunks = 6144 waves
    int mtile = idx / 12, chunk = idx % 12;
    int ln = lane & 15, half = lane >> 4;

    const float* arow = x + (mtile * 16 + ln) * D_EMB + half * 2;
    int ncol[4];
    v8f acc[4];
#pragma unroll
    for (int t = 0; t < 4; t++) {
        ncol[t] = chunk * 64 + t * 16 + ln;
        float bv = b[ncol[t]];
#pragma unroll
        for (int j = 0; j < 8; j++) acc[t][j] = bv;
    }
    for (int k = 0; k < D_EMB; k += 4) {
        v2f a; a.x = arow[k]; a.y = arow[k + 1];
#pragma unroll
        for (int t = 0; t < 4; t++) {
            const float* bp = W + (k + half * 2) * KQV_LD + ncol[t];
            v2f bb; bb.x = bp[0]; bb.y = bp[KQV_LD];
            acc[t] = WMMA_F32(a, bb, acc[t]);
        }
    }
    int rbase = mtile * 16 + half * 8;
#pragma unroll
    for (int t = 0; t < 4; t++)
#pragma unroll
        for (int j = 0; j < 8; j++)
            kqv[(rbase + j) * KQV_LD + ncol[t]] = acc[t][j];
}

// ---------------------------------------------------------------- 4) attention logits (WMMA)
__device__ __forceinline__ float feat_val(int f, const float* rowK, const float* rowQ,
                                          float ea0, float ea1, int h,
                                          const float* We, const float* be) {
    if (f < 32) return rowK[f] * KSCALE;          // k = kqv[src, D:2D]/sqrt(HD)
    if (f < 64) return rowQ[f - 32];              // q = kqv[dst, 0:D]
    int d = h * HD + (f - 64);                    // ef = relu(ea @ W_edge + b_edge)
    float v = fmaf(ea0, We[d], fmaf(ea1, We[D_EMB + d], be[d]));
    return v > 0.f ? v : 0.f;
}

// block = one group of 16 edges; wave w = head w. hdn[16x32] via WMMA, logit via LDS dot.
__global__ __launch_bounds__(256) void attn_logits(const float* __restrict__ kqv,
                                                   const int* __restrict__ src,
                                                   const int* __restrict__ dst,
                                                   const int* __restrict__ valid,
                                                   const float* __restrict__ edge_attr,
                                                   const float* __restrict__ W_edge,
                                                   const float* __restrict__ b_edge,
                                                   const float* __restrict__ W_att1,
                                                   const float* __restrict__ b_att1,
                                                   const float* __restrict__ W_att2,
                                                   const float* __restrict__ b_att2,
                                                   float* __restrict__ logits) {
    __shared__ float hdn_s[8][16][32];
    int lane = threadIdx.x & 31, h = threadIdx.x >> 5;
    int ln = lane & 15, half = lane >> 4;
    int group = blockIdx.x;
    int e = group * 16 + ln;

    const float* rowK = kqv + src[e] * KQV_LD + D_EMB + h * HD;
    const float* rowQ = kqv + dst[e] * KQV_LD + h * HD;
    float ea0 = 0.f, ea1 = 0.f;
    if (e < E_RAW)          { ea0 = edge_attr[2 * e];            ea1 = edge_attr[2 * e + 1]; }
    else if (e < 2 * E_RAW) { ea0 = edge_attr[2 * (e - E_RAW)];  ea1 = edge_attr[2 * (e - E_RAW) + 1]; }

    v8f acc0, acc1;
    {
        float b0 = b_att1[ln], b1 = b_att1[16 + ln];
#pragma unroll
        for (int j = 0; j < 8; j++) { acc0[j] = b0; acc1[j] = b1; }
    }
    for (int k = 0; k < 96; k += 4) {
        int f0 = k + half * 2;
        v2f a;
        a.x = feat_val(f0,     rowK, rowQ, ea0, ea1, h, W_edge, b_edge);
        a.y = feat_val(f0 + 1, rowK, rowQ, ea0, ea1, h, W_edge, b_edge);
        const float* bp = W_att1 + (k + half * 2) * HD;
        v2f bb0, bb1;
        bb0.x = bp[ln];      bb0.y = bp[HD + ln];
        bb1.x = bp[16 + ln]; bb1.y = bp[HD + 16 + ln];
        acc0 = WMMA_F32(a, bb0, acc0);
        acc1 = WMMA_F32(a, bb1, acc1);
    }
#pragma unroll
    for (int j = 0; j < 8; j++) {
        hdn_s[h][half * 8 + j][ln]      = fmaxf(acc0[j], 0.f);
        hdn_s[h][half * 8 + j][16 + ln] = fmaxf(acc1[j], 0.f);
    }
    __syncthreads();
    if (lane < 16) {
        float lg = b_att2[0];
#pragma unroll
        for (int n = 0; n < 32; n++) lg += hdn_s[h][lane][n] * W_att2[n];
        int ee = group * 16 + lane;
        logits[ee * H_HEADS + h] = valid[ee] ? lg : -1e30f;   // masked edges vanish in softmax
    }
}

// ---------------------------------------------------------------- 5) scatter softmax
__global__ void init_ms(float* __restrict__ m, float* __restrict__ s) {
    int i = blockIdx.x * blockDim.x + threadIdx.x;
    if (i < N_NODES * H_HEADS) { m[i] = -1e30f; s[i] = 0.f; }
}
__global__ void seg_max(const float* __restrict__ logits, const int* __restrict__ dst,
                        float* __restrict__ m) {
    int i = blockIdx.x * blockDim.x + threadIdx.x;
    if (i >= E2 * H_HEADS) return;
    int e = i >> 3, h = i & 7;
    atomicMaxF(&m[dst[e] * H_HEADS + h], logits[i]);
}
__global__ void seg_expsum(float* __restrict__ logits, const int* __restrict__ dst,
                           const float* __restrict__ m, float* __restrict__ s) {
    int i = blockIdx.x * blockDim.x + threadIdx.x;
    if (i >= E2 * H_HEADS) return;
    int e = i >> 3, h = i & 7;
    float a = expf(logits[i] - m[dst[e] * H_HEADS + h]);
    logits[i] = a;                              // buffer now holds numerators
    atomicAdd(&s[dst[e] * H_HEADS + h], a);
}
__global__ void norm_alpha(float* __restrict__ alpha, const int* __restrict__ dst,
                           const float* __restrict__ s) {
    int i = blockIdx.x * blockDim.x + threadIdx.x;
    if (i >= E2 * H_HEADS) return;
    int e = i >> 3, h = i & 7;
    alpha[i] = alpha[i] / (s[dst[e] * H_HEADS + h] + 1e-16f);
}

// ---------------------------------------------------------------- 6) message GEMM + scatter (WMMA)
// msg[16x256] = (alpha*v)[16x256] @ W_out[256x256]; wave handles 16 edges x 64 cols.
__global__ __launch_bounds__(256) void msg_gemm(const float* __restrict__ kqv,
                                                const float* __restrict__ alpha,
                                                const int* __restrict__ src,
                                                const int* __restrict__ dst,
                                                const int* __restrict__ valid,
                                                const float* __restrict__ W_out,
                                                const float* __restrict__ b_out,
                                                float* __restrict__ outbuf) {
    int lane = threadIdx.x & 31, wave = threadIdx.x >> 5;
    int idx = blockIdx.x * 8 + wave;           // (E2/16) groups * 4 chunks = 34816 waves
    int group = idx >> 2, chunk = idx & 3;
    int ln = lane & 15, half = lane >> 4;
    int e = group * 16 + ln;

    const float* rowV = kqv + src[e] * KQV_LD + 2 * D_EMB + half * 2;
    float al[8];
#pragma unroll
    for (int h = 0; h < 8; h++) al[h] = alpha[e * H_HEADS + h];

    int ncol[4];
    v8f acc[4];
#pragma unroll
    for (int t = 0; t < 4; t++) {
        ncol[t] = chunk * 64 + t * 16 + ln;
#pragma unroll
        for (int j = 0; j < 8; j++) acc[t][j] = 0.f;
    }
    for (int k = 0; k < D_EMB; k += 4) {
        float sc = al[k >> 5];                 // head = k/32 (k+half*2 stays in same head block)
        v2f a; a.x = sc * rowV[k]; a.y = sc * rowV[k + 1];
#pragma unroll
        for (int t = 0; t < 4; t++) {
            const float* bp = W_out + (k + half * 2) * D_EMB + ncol[t];
            v2f bb; bb.x = bp[0]; bb.y = bp[D_EMB];
            acc[t] = WMMA_F32(a, bb, acc[t]);
        }
    }
#pragma unroll
    for (int j = 0; j < 8; j++) {
        int er = group * 16 + half * 8 + j;
        if (valid[er]) {                        // invalid edges contribute nothing (incl. b_out)
            int de = dst[er];
#pragma unroll
            for (int t = 0; t < 4; t++)
                atomicAdd(&outbuf[de * D_EMB + ncol[t]], acc[t][j] + b_out[ncol[t]]);
        }
    }
}

// ---------------------------------------------------------------- 7) BatchNorm (3 stages)
__global__ __launch_bounds__(256) void bn_partial(const float* __restrict__ outbuf,
                                                  float* __restrict__ psum, float* __restrict__ psq) {
    int col = threadIdx.x, blk = blockIdx.x;   // 64 blocks * 128 rows, coalesced over cols
    float s = 0.f, q = 0.f;
    for (int r = blk * 128; r < blk * 128 + 128; r++) {
        float v = outbuf[r * D_EMB + col];
        s += v; q += v * v;
    }
    psum[blk * D_EMB + col] = s;
    psq[blk * D_EMB + col]  = q;
}
__global__ void bn_stats(const float* __restrict__ psum, const float* __restrict__ psq,
                         float* __restrict__ mean, float* __restrict__ varr) {
    int col = blockIdx.x * blockDim.x + threadIdx.x;
    if (col >= D_EMB) return;
    float s = 0.f, q = 0.f;
    for (int b = 0; b < 64; b++) { s += psum[b * D_EMB + col]; q += psq[b * D_EMB + col]; }
    float mu = s / (float)N_NODES;
    mean[col] = mu;
    varr[col] = q / (float)N_NODES - mu * mu;
}
__global__ void bn_apply(const float* __restrict__ outbuf, const float* __restrict__ mean,
                         const float* __restrict__ varr, const float* __restrict__ gamma,
                         const float* __restrict__ beta, float* __restrict__ out) {
    int i = blockIdx.x * blockDim.x + threadIdx.x;
    if (i >= N_NODES * D_EMB) return;
    int col = i & (D_EMB - 1);
    out[i] = (outbuf[i] - mean[col]) * rsqrtf(varr[col] + 1e-5f) * gamma[col] + beta[col];
}

// ---------------------------------------------------------------- launch
extern "C" void kernel_launch(void* const* d_in, const int* in_sizes, int n_in,
                              void* d_out, int out_size, void* d_ws, size_t ws_size,
                              hipStream_t stream) {
    const float*         x         = (const float*)d_in[0];
    const int*           edge_idx  = (const int*)d_in[1];
    const float*         edge_attr = (const float*)d_in[2];
    const unsigned char* qm        = (const unsigned char*)d_in[3];
    // d_in[4] = start_right (constant 4096, unused)
    const float* W_kqv  = (const float*)d_in[5];
    const float* b_kqv  = (const float*)d_in[6];
    const float* W_edge = (const float*)d_in[7];
    const float* b_edge = (const float*)d_in[8];
    const float* W_att1 = (const float*)d_in[9];
    const float* b_att1 = (const float*)d_in[10];
    const float* W_att2 = (const float*)d_in[11];
    const float* b_att2 = (const float*)d_in[12];
    const float* W_out  = (const float*)d_in[13];
    const float* b_out  = (const float*)d_in[14];
    const float* gamma  = (const float*)d_in[15];
    const float* beta   = (const float*)d_in[16];
    float* out = (float*)d_out;

    char* w = (char*)d_ws;
    auto alloc = [&](size_t bytes) -> char* {
        char* p = w; w += (bytes + 255) & ~(size_t)255; return p;
    };
    float* kqv    = (float*)alloc((size_t)N_NODES * KQV_LD * 4);  // 25.2 MB
    int*   src    = (int*)  alloc((size_t)E2 * 4);
    int*   dst    = (int*)  alloc((size_t)E2 * 4);
    int*   valid  = (int*)  alloc((size_t)E2 * 4);
    float* logits = (float*)alloc((size_t)E2 * H_HEADS * 4);      // 4.5 MB (logits->alpha)
    float* m      = (float*)alloc((size_t)N_NODES * H_HEADS * 4);
    float* s      = (float*)alloc((size_t)N_NODES * H_HEADS * 4);
    float* outbuf = (float*)alloc((size_t)N_NODES * D_EMB * 4);   // 8.4 MB
    float* psum   = (float*)alloc((size_t)64 * D_EMB * 4);
    float* psq    = (float*)alloc((size_t)64 * D_EMB * 4);
    float* meanb  = (float*)alloc((size_t)D_EMB * 4);
    float* varb   = (float*)alloc((size_t)D_EMB * 4);

    build_edges<<<(E2 + 255) / 256, 256, 0, stream>>>(edge_idx, qm, src, dst, valid);
    init_out<<<(N_NODES * D_EMB / 4 + 255) / 256, 256, 0, stream>>>((const float4*)x, (float4*)outbuf);

    kqv_gemm<<<(N_NODES / 16) * (KQV_LD / 64) / 8, 256, 0, stream>>>(x, W_kqv, b_kqv, kqv);

    attn_logits<<<E2 / 16, 256, 0, stream>>>(kqv, src, dst, valid, edge_attr,
                                             W_edge, b_edge, W_att1, b_att1, W_att2, b_att2, logits);

    init_ms<<<(N_NODES * H_HEADS + 255) / 256, 256, 0, stream>>>(m, s);
    seg_max<<<(E2 * H_HEADS + 255) / 256, 256, 0, stream>>>(logits, dst, m);
    seg_expsum<<<(E2 * H_HEADS + 255) / 256, 256, 0, stream>>>(logits, dst, m, s);
    norm_alpha<<<(E2 * H_HEADS + 255) / 256, 256, 0, stream>>>(logits, dst, s);

    msg_gemm<<<(E2 / 16) * 4 / 8, 256, 0, stream>>>(kqv, logits, src, dst, valid,
                                                    W_out, b_out, outbuf);

    bn_partial<<<64, 256, 0, stream>>>(outbuf, psum, psq);
    bn_stats<<<1, 256, 0, stream>>>(psum, psq, meanb, varb);
    bn_apply<<<(N_NODES * D_EMB + 255) / 256, 256, 0, stream>>>(outbuf, meanb, varb, gamma, beta, out);
}